// GraphAttentionLayer_52312701666008
// MI455X (gfx1250) — compile-verified
//
#include <hip/hip_runtime.h>
#include <hip/hip_bf16.h>
#include <math.h>

typedef float v2f __attribute__((ext_vector_type(2)));
typedef float v8f __attribute__((ext_vector_type(8)));

// Only check on the device pass — host clang never has amdgcn builtins.
#if defined(__HIP_DEVICE_COMPILE__) && !__has_builtin(__builtin_amdgcn_wmma_f32_16x16x4_f32)
#error "missing __builtin_amdgcn_wmma_f32_16x16x4_f32 on this toolchain (device pass)"
#endif

#define D 128          // D_IN == D_OUT
#define NHEAD 4
#define DH 32

// ---------- order-preserving float <-> uint key (for atomic max of floats) ----
__device__ __forceinline__ unsigned fkey(float f) {
  unsigned u = __float_as_uint(f);
  return (u & 0x80000000u) ? ~u : (u | 0x80000000u);
}
__device__ __forceinline__ float keyf(unsigned k) {
  unsigned u = (k & 0x80000000u) ? (k & 0x7FFFFFFFu) : ~k;
  return __uint_as_float(u);
}
__device__ __forceinline__ float lrelu(float x) { return x > 0.f ? x : 0.2f * x; }

// ---------------------------------------------------------------------------
// K1: h[N,128] = x[N,128] @ Wcat[128,128] via V_WMMA_F32_16X16X4_F32 (fp32).
// Block = 256 threads = 8 waves; wave w -> 16-col tile, block -> 16-row stripe.
// A frag (16x4 f32): lane<16 holds M=lane, K pair {k,k+1}; lane>=16 K pair {k+2,k+3}.
// B frag (4x16 f32): lane<16 holds N=lane, K pair {k,k+1}; lane>=16 K pair {k+2,k+3}.
// D (16x16 f32): vgpr r, lane<16 -> (M=r, N=lane); lane>=16 -> (M=r+8, N=lane-16).
// ---------------------------------------------------------------------------
__global__ void gat_gemm_wmma(const float* __restrict__ x,
                              const float* __restrict__ W, // [H,128,32]
                              float* __restrict__ h, int N) {
#if defined(__HIP_DEVICE_COMPILE__)
  const int wave = threadIdx.x >> 5;   // 0..7 : column tile
  const int lane = threadIdx.x & 31;
  const int r0 = blockIdx.x * 16;
  if (r0 >= N) return;                 // uniform across block -> EXEC stays full
  const int m    = lane & 15;          // row (A) / col (B) within tile
  const int hi   = lane >> 4;          // lane-half selects K pair
  const int n0   = wave * 16;          // output column base
  const int head = n0 >> 5;
  const int f0   = n0 & 31;
  int arow = r0 + m; if (arow >= N) arow = N - 1;          // safe clamp (no divergence)
  const float* xr = x + (size_t)arow * D + hi * 2;         // A: contiguous K pair
  const float* Wb = W + (size_t)head * (D * DH) + f0 + m;  // B: stride 32 between K's

  v8f c = {};
#pragma unroll
  for (int kb = 0; kb < 32; ++kb) {
    const int k = kb * 4;
    v2f a = *(const v2f*)(xr + k);                          // x[arow][k+2hi], [k+2hi+1]
    v2f b;
    b.x = Wb[(size_t)(k + hi * 2) * DH];                    // Wcat[k+2hi][n0+m]
    b.y = Wb[(size_t)(k + hi * 2 + 1) * DH];                // Wcat[k+2hi+1][n0+m]
    c = __builtin_amdgcn_wmma_f32_16x16x4_f32(false, a, false, b, (short)0, c,
                                              false, false);
  }
#pragma unroll
  for (int r = 0; r < 8; ++r) {
    const int row = r0 + r + hi * 8;
    if (row < N) h[(size_t)row * D + n0 + m] = c[r];
  }
#endif
}

// K2: per-(node,head) attention dots; also init max-keys and sums --------------
__global__ void gat_att(const float* __restrict__ h,
                        const float* __restrict__ att_src,
                        const float* __restrict__ att_dst,
                        float* __restrict__ a_src, float* __restrict__ a_dst,
                        unsigned* __restrict__ mkey, float* __restrict__ ssum,
                        int NH) {
  const int t = blockIdx.x * blockDim.x + threadIdx.x;
  if (t >= NH) return;
  const int n = t >> 2, hd = t & 3;
  const float* hp = h + (size_t)n * D + hd * DH;
  const float* as = att_src + hd * DH;
  const float* ad = att_dst + hd * DH;
  float s0 = 0.f, s1 = 0.f;
#pragma unroll
  for (int f = 0; f < DH; ++f) { const float v = hp[f]; s0 += v * as[f]; s1 += v * ad[f]; }
  a_src[t] = s0; a_dst[t] = s1;
  mkey[t] = 0u;        // key(-inf) lower bound
  ssum[t] = 0.f;
}

// K3: segment max of LeakyReLU logits via uint atomicMax ----------------------
__global__ void gat_edge_max(const int* __restrict__ ei,
                             const float* __restrict__ a_src,
                             const float* __restrict__ a_dst,
                             unsigned* __restrict__ mkey, int E, int Et) {
  const int e = blockIdx.x * blockDim.x + threadIdx.x;
  if (e >= Et) return;
  int s, d;
  if (e < E) { s = ei[e]; d = ei[E + e]; } else { s = d = e - E; }
#pragma unroll
  for (int hd = 0; hd < NHEAD; ++hd) {
    const float l = lrelu(a_src[s * 4 + hd] + a_dst[d * 4 + hd]);
    atomicMax(&mkey[d * 4 + hd], fkey(l));
  }
}

// K4: w = exp(l - max); stash per-edge w, atomicAdd segment sums --------------
__global__ void gat_edge_expsum(const int* __restrict__ ei,
                                const float* __restrict__ a_src,
                                const float* __restrict__ a_dst,
                                const unsigned* __restrict__ mkey,
                                float* __restrict__ ssum,
                                float* __restrict__ ealpha, int E, int Et) {
  const int e = blockIdx.x * blockDim.x + threadIdx.x;
  if (e >= Et) return;
  int s, d;
  if (e < E) { s = ei[e]; d = ei[E + e]; } else { s = d = e - E; }
#pragma unroll
  for (int hd = 0; hd < NHEAD; ++hd) {
    const float l = lrelu(a_src[s * 4 + hd] + a_dst[d * 4 + hd]);
    const float w = expf(l - keyf(mkey[d * 4 + hd]));
    ealpha[(size_t)e * 4 + hd] = w;
    atomicAdd(&ssum[d * 4 + hd], w);
  }
}

// K5: out = bias --------------------------------------------------------------
__global__ void gat_init_out(float* __restrict__ out,
                             const float* __restrict__ bias, size_t total) {
  const size_t t = (size_t)blockIdx.x * blockDim.x + threadIdx.x;
  if (t < total) out[t] = bias[t & (D - 1)];
}

// K6: one wave per edge; lane owns 4 output components; float4 gather + atomics
__global__ void gat_scatter(const int* __restrict__ ei,
                            const float* __restrict__ h,
                            const float* __restrict__ ealpha,
                            const float* __restrict__ ssum,
                            float* __restrict__ out, int E, int Et) {
  const int lane = threadIdx.x & 31;
  const int wave = blockIdx.x * (blockDim.x >> 5) + (threadIdx.x >> 5);
  const int nwaves = gridDim.x * (blockDim.x >> 5);
  const int hd = lane >> 3;                    // head = (4*lane)/32
  for (int e = wave; e < Et; e += nwaves) {
    int s, d;
    if (e < E) { s = ei[e]; d = ei[E + e]; } else { s = d = e - E; }
    const float alpha = ealpha[(size_t)e * 4 + hd] / (ssum[d * 4 + hd] + 1e-16f);
    const float4 hv = *(const float4*)(h + (size_t)s * D + lane * 4);
    float* op = out + (size_t)d * D + lane * 4;
    atomicAdd(op + 0, hv.x * alpha);
    atomicAdd(op + 1, hv.y * alpha);
    atomicAdd(op + 2, hv.z * alpha);
    atomicAdd(op + 3, hv.w * alpha);
  }
}

// K7: deterministic per-column mean / biased var ------------------------------
__global__ void gat_colstats(const float* __restrict__ out,
                             float* __restrict__ cmean, float* __restrict__ cvar,
                             int N) {
  __shared__ float sh[256], sq[256];
  const int c = blockIdx.x;         // 0..127
  const int t = threadIdx.x;
  float s = 0.f, q = 0.f;
  for (int r = t; r < N; r += 256) {
    const float v = out[(size_t)r * D + c];
    s += v; q += v * v;
  }
  sh[t] = s; sq[t] = q; __syncthreads();
  for (int o = 128; o > 0; o >>= 1) {
    if (t < o) { sh[t] += sh[t + o]; sq[t] += sq[t + o]; }
    __syncthreads();
  }
  if (t == 0) {
    const float mu = sh[0] / (float)N;
    cmean[c] = mu;
    cvar[c] = sq[0] / (float)N - mu * mu;
  }
}

// K8: in-place BN apply -------------------------------------------------------
__global__ void gat_bn(float* __restrict__ out,
                       const float* __restrict__ cmean, const float* __restrict__ cvar,
                       const float* __restrict__ gamma, const float* __restrict__ beta,
                       size_t total) {
  const size_t t = (size_t)blockIdx.x * blockDim.x + threadIdx.x;
  if (t >= total) return;
  const int c = (int)(t & (D - 1));
  out[t] = (out[t] - cmean[c]) * rsqrtf(cvar[c] + 1e-5f) * gamma[c] + beta[c];
}

extern "C" void kernel_launch(void* const* d_in, const int* in_sizes, int n_in,
                              void* d_out, int out_size, void* d_ws, size_t ws_size,
                              hipStream_t stream) {
  const float* x       = (const float*)d_in[0];
  const int*   ei      = (const int*)d_in[1];
  const float* W       = (const float*)d_in[2];
  const float* att_src = (const float*)d_in[3];
  const float* att_dst = (const float*)d_in[4];
  const float* bias    = (const float*)d_in[5];
  const float* gamma   = (const float*)d_in[6];
  const float* beta    = (const float*)d_in[7];

  const int N  = in_sizes[0] / D;
  const int E  = in_sizes[1] / 2;
  const int Et = E + N;
  const int NH = N * NHEAD;
  const size_t total = (size_t)N * D;

  // workspace carve-up (all 4-byte types)
  float*    h      = (float*)d_ws;                       // N*128
  float*    a_src  = h + total;                          // N*4
  float*    a_dst  = a_src + (size_t)NH;                 // N*4
  unsigned* mkey   = (unsigned*)(a_dst + (size_t)NH);    // N*4
  float*    ssum   = (float*)(mkey + (size_t)NH);        // N*4
  float*    ealpha = ssum + (size_t)NH;                  // Et*4
  float*    cmean  = ealpha + (size_t)Et * NHEAD;        // 128
  float*    cvar   = cmean + D;                          // 128
  float*    out    = (float*)d_out;

  gat_gemm_wmma<<<(N + 15) / 16, 256, 0, stream>>>(x, W, h, N);
  gat_att<<<(NH + 255) / 256, 256, 0, stream>>>(h, att_src, att_dst,
                                                a_src, a_dst, mkey, ssum, NH);
  gat_edge_max<<<(Et + 255) / 256, 256, 0, stream>>>(ei, a_src, a_dst, mkey, E, Et);
  gat_edge_expsum<<<(Et + 255) / 256, 256, 0, stream>>>(ei, a_src, a_dst, mkey,
                                                        ssum, ealpha, E, Et);
  gat_init_out<<<(int)((total + 255) / 256), 256, 0, stream>>>(out, bias, total);
  gat_scatter<<<4096, 256, 0, stream>>>(ei, h, ealpha, ssum, out, E, Et);
  gat_colstats<<<D, 256, 0, stream>>>(out, cmean, cvar, N);
  gat_bn<<<(int)((total + 255) / 256), 256, 0, stream>>>(out, cmean, cvar,
                                                         gamma, beta, total);
}